// Model_31756988187003
// MI455X (gfx1250) — compile-verified
//
#include <hip/hip_runtime.h>
#include <cstdint>
#include <cstddef>

// Problem constants (from reference)
#define B_     4096
#define F_     4096
#define V_     1024
#define E_     512
#define H_     1024
#define L_     30
#define BOUND_ 1023
#define NDIS_  3
#define VLW_   0.1f
#define BW_    0.1f

typedef float v2f __attribute__((ext_vector_type(2)));
typedef float v8f __attribute__((ext_vector_type(8)));

__device__ __forceinline__ float sigmoidf_(float x) { return 1.0f / (1.0f + __expf(-x)); }

// ---------------------------------------------------------------------------
// FP32 WMMA NT-GEMM: C[M,N] = A[M,K] * W[N,K]^T (+bias1+bias2) (+C if accum)
// Block tile 128x128, K-chunk 32, 8 waves (4M x 2N); each wave owns a 32x64
// tile = 2x4 V_WMMA_F32_16X16X4_F32 accumulators (8 WMMA per k-step against
// 6 ds_load_2addr_b64 -> better matrix-pipe:LDS ratio than 32x32).
// Register-staged software pipeline: stage k+1 global loads are issued before
// the 64-WMMA compute phase of stage k, hiding L2 latency behind the XDL work.
// ---------------------------------------------------------------------------
#define BM 128
#define BN 128
#define BK 32
#define LDSP (BK + 4)   // row stride 36 floats = 144B -> keeps b128/b64 aligned

__global__ __launch_bounds__(256) void gemm_nt_wmma(
    const float* __restrict__ A, const float* __restrict__ W,
    const float* __restrict__ bias1, const float* __restrict__ bias2,
    float* __restrict__ C, int M, int N, int K, int accum)
{
  __shared__ float As[BM][LDSP];
  __shared__ float Bs[BN][LDSP];

  const int tid  = threadIdx.x;
  const int lane = tid & 31;
  const int wave = tid >> 5;
  const int wm   = (wave & 3) * 32;   // wave row offset inside block tile (4 waves in M)
  const int wn   = (wave >> 2) * 64;  // wave col offset inside block tile (2 waves in N)
  const int half = lane >> 4;
  const int l16  = lane & 15;
  const int bM   = blockIdx.y * BM;
  const int bN   = blockIdx.x * BN;

  // Per-thread staging slots: 4 float4 from each of A and B tiles (128x32 each)
  const int srow = tid >> 3;          // tile row for i=0 slot (tid+i*256 below)
  const int sc4  = (tid & 7) << 2;    // float offset within the 32-wide K chunk

  float4 ra[4], rb[4];
#pragma unroll
  for (int i = 0; i < 4; ++i) {
    int r = srow + i * 32;
    ra[i] = *(const float4*)&A[(size_t)(bM + r) * K + sc4];
    rb[i] = *(const float4*)&W[(size_t)(bN + r) * K + sc4];
  }

  v8f acc[2][4] = {};

  for (int k0 = 0; k0 < K; k0 += BK) {
    // Commit staged registers to LDS (compiler inserts s_wait_loadcnt here,
    // which is the only point the previous stage's global loads must be done).
#pragma unroll
    for (int i = 0; i < 4; ++i) {
      int r = srow + i * 32;
      *(float4*)&As[r][sc4] = ra[i];
      *(float4*)&Bs[r][sc4] = rb[i];
    }
    __syncthreads();

    // Issue next stage's global loads; they overlap the WMMA phase below.
    const int kn = k0 + BK;
    if (kn < K) {
#pragma unroll
      for (int i = 0; i < 4; ++i) {
        int r = srow + i * 32;
        ra[i] = *(const float4*)&A[(size_t)(bM + r) * K + kn + sc4];
        rb[i] = *(const float4*)&W[(size_t)(bN + r) * K + kn + sc4];
      }
    }

#pragma unroll
    for (int ks = 0; ks < BK / 4; ++ks) {
      const int kk = ks * 4 + half * 2;  // frag: lanes 0-15 K={0,1}, 16-31 K={2,3}
      v2f a0 = *(const v2f*)&As[wm + l16][kk];
      v2f a1 = *(const v2f*)&As[wm + 16 + l16][kk];
      v2f b0 = *(const v2f*)&Bs[wn + l16][kk];
      v2f b1 = *(const v2f*)&Bs[wn + 16 + l16][kk];
      v2f b2 = *(const v2f*)&Bs[wn + 32 + l16][kk];
      v2f b3 = *(const v2f*)&Bs[wn + 48 + l16][kk];
      acc[0][0] = __builtin_amdgcn_wmma_f32_16x16x4_f32(false, a0, false, b0, (short)0, acc[0][0], false, false);
      acc[0][1] = __builtin_amdgcn_wmma_f32_16x16x4_f32(false, a0, false, b1, (short)0, acc[0][1], false, false);
      acc[0][2] = __builtin_amdgcn_wmma_f32_16x16x4_f32(false, a0, false, b2, (short)0, acc[0][2], false, false);
      acc[0][3] = __builtin_amdgcn_wmma_f32_16x16x4_f32(false, a0, false, b3, (short)0, acc[0][3], false, false);
      acc[1][0] = __builtin_amdgcn_wmma_f32_16x16x4_f32(false, a1, false, b0, (short)0, acc[1][0], false, false);
      acc[1][1] = __builtin_amdgcn_wmma_f32_16x16x4_f32(false, a1, false, b1, (short)0, acc[1][1], false, false);
      acc[1][2] = __builtin_amdgcn_wmma_f32_16x16x4_f32(false, a1, false, b2, (short)0, acc[1][2], false, false);
      acc[1][3] = __builtin_amdgcn_wmma_f32_16x16x4_f32(false, a1, false, b3, (short)0, acc[1][3], false, false);
    }
    __syncthreads();
  }

  // Epilogue. C frag layout: VGPR r -> row = r + 8*half, col = l16.
#pragma unroll
  for (int mi = 0; mi < 2; ++mi) {
#pragma unroll
    for (int ni = 0; ni < 4; ++ni) {
      int col = bN + wn + ni * 16 + l16;
      float bb = 0.0f;
      if (bias1) bb += bias1[col];
      if (bias2) bb += bias2[col];
#pragma unroll
      for (int r = 0; r < 8; ++r) {
        int row = bM + wm + mi * 16 + half * 8 + r;
        size_t off = (size_t)row * N + col;
        float v = acc[mi][ni][r] + bb;
        if (accum) v += C[off];
        C[off] = v;
      }
    }
  }
}

// ---------------------------------------------------------------------------
// nwc = wc / sum(wc) with wc[BOUND] *= BW (sum over all V since BOUND==V-1)
// ---------------------------------------------------------------------------
__global__ __launch_bounds__(1024) void prep_nwc(const float* __restrict__ wc,
                                                 float* __restrict__ nwc)
{
  __shared__ float red[1024];
  int t = threadIdx.x;
  float v = wc[t];
  if (t == BOUND_) v *= BW_;
  red[t] = v;
  __syncthreads();
  for (int s = 512; s > 0; s >>= 1) {
    if (t < s) red[t] += red[t + s];
    __syncthreads();
  }
  float denom = red[0];
  nwc[t] = (denom > 0.0f) ? v / denom : v;
}

__global__ void fill_f32(float* __restrict__ p, float v, int n)
{
  int i = blockIdx.x * 256 + threadIdx.x;
  if (i < n) p[i] = v;
}

__global__ void init_sender(int* __restrict__ tok, int* __restrict__ sl,
                            float* __restrict__ vl, int* __restrict__ m)
{
  int b = blockIdx.x * 256 + threadIdx.x;
  tok[b] = BOUND_;
  sl[b]  = L_ + 1;
  vl[b]  = 0.0f;
  m[b * (L_ + 1)] = BOUND_;
}

// out[b, :] = table[toks[b*stride + off], :]   (vectorized float4, E=512)
__global__ __launch_bounds__(256) void gather_emb(
    const float4* __restrict__ table, const int* __restrict__ toks,
    int tokStride, int tokOffset, float4* __restrict__ out)
{
  int idx = blockIdx.x * 256 + threadIdx.x;   // B*E/4
  int b = idx >> 7;                           // E/4 == 128
  int e = idx & 127;
  int tk = toks[b * tokStride + tokOffset];
  out[idx] = table[(size_t)tk * (E_ / 4) + e];
}

// LSTM pointwise: c = sig(f)*c + sig(i)*tanh(g); h = sig(o)*tanh(c)
__global__ __launch_bounds__(256) void lstm_update(
    const float* __restrict__ gates, float* __restrict__ h, float* __restrict__ c)
{
  int idx = blockIdx.x * 256 + threadIdx.x;   // B*H
  int b = idx >> 10;                          // H == 1024
  int j = idx & 1023;
  const float* g = gates + (size_t)b * (4 * H_);
  float gi = g[j];
  float gf = g[H_ + j];
  float gg = g[2 * H_ + j];
  float go = g[3 * H_ + j];
  float cc = sigmoidf_(gf) * c[idx] + sigmoidf_(gi) * tanhf(gg);
  c[idx] = cc;
  h[idx] = sigmoidf_(go) * tanhf(cc);
}

// Per-row (block per b): token = argmax(scores) (first max wins like jnp.argmax),
// ce = logsumexp(scores - nwc) - (scores-nwc)[token]; update tok/sl/vl/m.
__global__ __launch_bounds__(256) void sender_select(
    const float* __restrict__ scores, const float* __restrict__ nwc,
    int* __restrict__ tok, int* __restrict__ sl, float* __restrict__ vl,
    int* __restrict__ m, int step)
{
  __shared__ float sf[256];
  __shared__ int   si[256];
  const int b = blockIdx.x, t = threadIdx.x;
  const float* row = scores + (size_t)b * V_;

  // argmax (strict > keeps lowest index within a thread's ascending stride)
  float best = -INFINITY; int bidx = V_;
  for (int j = t; j < V_; j += 256) {
    float v = row[j];
    if (v > best) { best = v; bidx = j; }
  }
  sf[t] = best; si[t] = bidx;
  __syncthreads();
  for (int s = 128; s > 0; s >>= 1) {
    if (t < s) {
      float v2 = sf[t + s]; int i2 = si[t + s];
      if (v2 > sf[t] || (v2 == sf[t] && i2 < si[t])) { sf[t] = v2; si[t] = i2; }
    }
    __syncthreads();
  }
  const int token = si[0];
  __syncthreads();

  // logsumexp over logits = scores - nwc
  float lmax = -INFINITY;
  for (int j = t; j < V_; j += 256) lmax = fmaxf(lmax, row[j] - nwc[j]);
  sf[t] = lmax;
  __syncthreads();
  for (int s = 128; s > 0; s >>= 1) {
    if (t < s) sf[t] = fmaxf(sf[t], sf[t + s]);
    __syncthreads();
  }
  const float lm = sf[0];
  __syncthreads();

  float se = 0.0f;
  for (int j = t; j < V_; j += 256) se += __expf(row[j] - nwc[j] - lm);
  sf[t] = se;
  __syncthreads();
  for (int s = 128; s > 0; s >>= 1) {
    if (t < s) sf[t] += sf[t + s];
    __syncthreads();
  }

  if (t == 0) {
    float lse = lm + __logf(sf[0]);
    vl[b] += lse - (row[token] - nwc[token]);
    tok[b] = token;
    if (token == BOUND_ && sl[b] == L_ + 1) sl[b] = step + 2;
    m[b * (L_ + 1) + step + 1] = token;
  }
}

// Pad message past seq length with BOUND; emit float m to output; histogram.
__global__ void pad_hist(int* __restrict__ m, const int* __restrict__ sl,
                         float* __restrict__ m_out, float* __restrict__ wc)
{
  int idx = blockIdx.x * 256 + threadIdx.x;   // B*(L+1)
  int b   = idx / (L_ + 1);
  int pos = idx % (L_ + 1);
  int v = (pos >= sl[b]) ? BOUND_ : m[idx];
  m[idx] = v;
  m_out[idx] = (float)v;
  atomicAdd(&wc[v], 1.0f);  // integer-valued adds: exact & order-independent
}

// Block per b: 4 dot products over F, hinge loss + accuracy.
__global__ __launch_bounds__(256) void score_hinge(
    const float* __restrict__ target, const float* __restrict__ dis,
    const float* __restrict__ r, const float* __restrict__ vl,
    float* __restrict__ lossb, float* __restrict__ accb)
{
  __shared__ float4 red[256];
  const int b = blockIdx.x, t = threadIdx.x;
  const float* rb = r + (size_t)b * F_;
  const float* tb = target + (size_t)b * F_;
  const float* d0p = dis + (size_t)b * F_;
  const float* d1p = dis + (size_t)B_ * F_ + (size_t)b * F_;
  const float* d2p = dis + 2 * (size_t)B_ * F_ + (size_t)b * F_;
  float4 s = {0.f, 0.f, 0.f, 0.f};
  for (int f = t; f < F_; f += 256) {
    float rv = rb[f];
    s.x += tb[f] * rv;
    s.y += d0p[f] * rv;
    s.z += d1p[f] * rv;
    s.w += d2p[f] * rv;
  }
  red[t] = s;
  __syncthreads();
  for (int st = 128; st > 0; st >>= 1) {
    if (t < st) {
      red[t].x += red[t + st].x; red[t].y += red[t + st].y;
      red[t].z += red[t + st].z; red[t].w += red[t + st].w;
    }
    __syncthreads();
  }
  if (t == 0) {
    float ts = red[0].x, d0 = red[0].y, d1 = red[0].z, d2 = red[0].w;
    float loss = fmaxf(0.f, 1.f - ts + d0)
               + fmaxf(0.f, 1.f - ts + d1)
               + fmaxf(0.f, 1.f - ts + d2);
    lossb[b] = loss + VLW_ * vl[b];
    // argmax(exp([ts,d0,d1,d2])) == 0  <=>  ts >= all distractor scores
    accb[b] = (ts >= d0 && ts >= d1 && ts >= d2) ? 1.0f : 0.0f;
  }
}

__global__ __launch_bounds__(256) void final_reduce(
    const float* __restrict__ lossb, const float* __restrict__ accb,
    float* __restrict__ out)
{
  __shared__ float rl[256], ra[256];
  int t = threadIdx.x;
  float ls = 0.f, ac = 0.f;
  for (int i = t; i < B_; i += 256) { ls += lossb[i]; ac += accb[i]; }
  rl[t] = ls; ra[t] = ac;
  __syncthreads();
  for (int s = 128; s > 0; s >>= 1) {
    if (t < s) { rl[t] += rl[t + s]; ra[t] += ra[t + s]; }
    __syncthreads();
  }
  if (t == 0) { out[0] = rl[0] / (float)B_; out[1] = ra[0] / (float)B_; }
}

// ---------------------------------------------------------------------------
extern "C" void kernel_launch(void* const* d_in, const int* in_sizes, int n_in,
                              void* d_out, int out_size, void* d_ws, size_t ws_size,
                              hipStream_t stream)
{
  (void)in_sizes; (void)n_in; (void)out_size; (void)ws_size;

  const float* target = (const float*)d_in[0];
  const float* dis    = (const float*)d_in[1];
  const float* wc_in  = (const float*)d_in[2];
  const float* emb_s  = (const float*)d_in[3];
  const float* affsW  = (const float*)d_in[4];
  const float* affsb  = (const float*)d_in[5];
  const float* Wih    = (const float*)d_in[6];
  const float* Whh    = (const float*)d_in[7];
  const float* bih    = (const float*)d_in[8];
  const float* bhh    = (const float*)d_in[9];
  const float* lpW    = (const float*)d_in[10];
  const float* lpb    = (const float*)d_in[11];
  const float* emb_r  = (const float*)d_in[12];
  const float* rWih   = (const float*)d_in[13];
  const float* rWhh   = (const float*)d_in[14];
  const float* rbih   = (const float*)d_in[15];
  const float* rbhh   = (const float*)d_in[16];
  const float* affrW  = (const float*)d_in[17];
  const float* affrb  = (const float*)d_in[18];

  float* out    = (float*)d_out;
  float* out_m  = out + 2;                       // [B, L+1] as float
  float* out_wc = out + 2 + B_ * (L_ + 1);       // [V]

  // Workspace carve-up (~127 MB)
  char* w = (char*)d_ws;
  auto allocF = [&](size_t n) { float* p = (float*)w; w += n * sizeof(float); return p; };
  float* nwc    = allocF(V_);
  float* h      = allocF((size_t)B_ * H_);
  float* c      = allocF((size_t)B_ * H_);
  float* emb    = allocF((size_t)B_ * E_);
  float* gates  = allocF((size_t)B_ * 4 * H_);   // reused as r at the end
  float* scores = allocF((size_t)B_ * V_);
  float* vl     = allocF(B_);
  float* lossb  = allocF(B_);
  float* accb   = allocF(B_);
  int* tok = (int*)w; w += (size_t)B_ * sizeof(int);
  int* sl  = (int*)w; w += (size_t)B_ * sizeof(int);
  int* mi  = (int*)w; w += (size_t)B_ * (L_ + 1) * sizeof(int);

  auto gemm = [&](const float* A, const float* Wm, const float* b1, const float* b2,
                  float* C, int M, int N, int K, int accum) {
    dim3 grid(N / BN, M / BM);
    gemm_nt_wmma<<<grid, 256, 0, stream>>>(A, Wm, b1, b2, C, M, N, K, accum);
  };

  // ---- setup ----
  prep_nwc<<<1, 1024, 0, stream>>>(wc_in, nwc);
  gemm(target, affsW, affsb, nullptr, h, B_, H_, F_, 0);        // h0
  fill_f32<<<(B_ * H_) / 256, 256, 0, stream>>>(c, 0.0f, B_ * H_);
  init_sender<<<B_ / 256, 256, 0, stream>>>(tok, sl, vl, mi);

  // ---- sender greedy decode ----
  for (int i = 0; i < L_; ++i) {
    gather_emb<<<(B_ * E_ / 4) / 256, 256, 0, stream>>>((const float4*)emb_s, tok, 1, 0, (float4*)emb);
    gemm(emb, Wih, bih, bhh, gates, B_, 4 * H_, E_, 0);
    gemm(h, Whh, nullptr, nullptr, gates, B_, 4 * H_, H_, 1);
    lstm_update<<<(B_ * H_) / 256, 256, 0, stream>>>(gates, h, c);
    gemm(h, lpW, lpb, nullptr, scores, B_, V_, H_, 0);
    sender_select<<<B_, 256, 0, stream>>>(scores, nwc, tok, sl, vl, mi, i);
  }

  // ---- pad + histogram ----
  fill_f32<<<V_ / 256, 256, 0, stream>>>(out_wc, 0.0f, V_);
  pad_hist<<<(B_ * (L_ + 1)) / 256, 256, 0, stream>>>(mi, sl, out_m, out_wc);

  // ---- receiver (reuse h/c buffers, zeroed) ----
  fill_f32<<<(B_ * H_) / 256, 256, 0, stream>>>(h, 0.0f, B_ * H_);
  fill_f32<<<(B_ * H_) / 256, 256, 0, stream>>>(c, 0.0f, B_ * H_);
  for (int t = 0; t < L_ + 1; ++t) {
    gather_emb<<<(B_ * E_ / 4) / 256, 256, 0, stream>>>((const float4*)emb_r, mi, L_ + 1, t, (float4*)emb);
    gemm(emb, rWih, rbih, rbhh, gates, B_, 4 * H_, E_, 0);
    gemm(h, rWhh, nullptr, nullptr, gates, B_, 4 * H_, H_, 1);
    lstm_update<<<(B_ * H_) / 256, 256, 0, stream>>>(gates, h, c);
  }

  // ---- final projection + scoring ----
  float* r = gates;  // gates buffer dead now; reuse for [B, F]
  gemm(h, affrW, affrb, nullptr, r, B_, F_, H_, 0);
  score_hinge<<<B_, 256, 0, stream>>>(target, dis, r, vl, lossb, accb);
  final_reduce<<<1, 256, 0, stream>>>(lossb, accb, out);
}